// SimpleAttention_343597384489
// MI455X (gfx1250) — compile-verified
//
#include <hip/hip_runtime.h>

// SimpleAttention collapses algebraically: softmax is over axis=1 (query axis)
// and the final mean is over that same axis, so sum_q att[b,q,s] == 1 and
//   mean_q (att @ v)[b,q,k] = (1/S) * sum_s v[b,s,k] = xbar[b,:] @ Wv + bv
// where xbar = mean over sequence of x. Q/K/scores/softmax are dead code.
//
// Plan:
//   K1: partial column sums of x over s   (64 MB stream -> bandwidth bound)
//   K2: finalize zero-PADDED xbar [16 x 1024] (rows 8..15 = 0) -> branchless K3
//   K3: [16 x 1024] @ [1024 x 512] GEMM with V_WMMA_F32_16X16X4_F32

#define B_ 8
#define BPAD_ 16
#define S_ 2048
#define D_ 1024
#define DK_ 512
#define NCHUNK 64
#define RPC (S_ / NCHUNK) // 32 rows per chunk

typedef __attribute__((ext_vector_type(2))) float v2f;
typedef __attribute__((ext_vector_type(8))) float v8f;

// ---------------- K1: partial column sums over s ----------------
// grid: B*NCHUNK blocks, 256 threads; thread t handles floats [4t, 4t+3] of D.
__global__ void colsum_partial(const float* __restrict__ x,
                               float* __restrict__ partials) {
    const int b     = blockIdx.x / NCHUNK;
    const int chunk = blockIdx.x % NCHUNK;
    const int t     = threadIdx.x; // 0..255 -> float4 lane over D/4 = 256

    const float4* __restrict__ xr = reinterpret_cast<const float4*>(x);
    float4 acc = make_float4(0.f, 0.f, 0.f, 0.f);
    size_t base = ((size_t)b * S_ + (size_t)chunk * RPC) * (D_ / 4) + t;
#pragma unroll 8
    for (int r = 0; r < RPC; ++r) {
        float4 v = xr[base + (size_t)r * (D_ / 4)];
        acc.x += v.x; acc.y += v.y; acc.z += v.z; acc.w += v.w;
    }
    // layout: partials[chunk][b][d]
    float4* __restrict__ p = reinterpret_cast<float4*>(partials);
    p[(size_t)(chunk * B_ + b) * (D_ / 4) + t] = acc;
}

// ---------------- K2: finalize PADDED xbar (rows >= 8 are zero) --------------
// grid covers BPAD_*D_ = 16384 entries.
__global__ void colsum_finalize_pad(const float* __restrict__ partials,
                                    float* __restrict__ xbarp) {
    const int col = blockIdx.x * blockDim.x + threadIdx.x; // 0..BPAD_*D_-1
    const int b   = col / D_;
    float r = 0.f;
    if (b < B_) {
        float acc = 0.f;
#pragma unroll 8
        for (int c = 0; c < NCHUNK; ++c)
            acc += partials[(size_t)c * (B_ * D_) + col];
        r = acc * (1.0f / (float)S_);
    }
    xbarp[col] = r;
}

// ---------------- K3: out = xbarp @ Wv + bv via V_WMMA_F32_16X16X4_F32 -------
// One wave (32 threads) per 16-wide N tile. Branchless inner loop.
// A frag (16x4 f32, 2 VGPRs): lane: M=lane&15, K = sel, sel+1 (sel = lane<16?0:2)
// B frag (4x16 f32, 2 VGPRs): lane: N=lane&15, same K pattern.
// C/D (16x16 f32, 8 VGPRs): VGPR r, lanes 0-15 -> (M=r, N=lane).
__global__ void wmma_gemm_out(const float* __restrict__ xbarp,
                              const float* __restrict__ Wv,
                              const float* __restrict__ bv,
                              float* __restrict__ out) {
    const int lane  = threadIdx.x;      // 0..31 (wave32)
    const int ntile = blockIdx.x;       // 0..31
    const int m     = lane & 15;
    const int sel   = (lane < 16) ? 0 : 2;
    const int n     = ntile * 16 + m;   // B-frag / store column
    const float* __restrict__ arow = xbarp + (size_t)m * D_;

    v8f c = {0.f, 0.f, 0.f, 0.f, 0.f, 0.f, 0.f, 0.f};

#pragma unroll 8
    for (int k0 = 0; k0 < D_; k0 += 4) {
        const int ka = k0 + sel;               // even -> 8B-aligned b64 load
        v2f a = *reinterpret_cast<const v2f*>(arow + ka);
        v2f bf;
        bf.x = Wv[(size_t)ka * DK_ + n];       // lanes 0-15: 64B coalesced
        bf.y = Wv[(size_t)(ka + 1) * DK_ + n];
        // (neg_a, A, neg_b, B, c_mod, C, reuse_a, reuse_b)
        c = __builtin_amdgcn_wmma_f32_16x16x4_f32(
                false, a, false, bf, (short)0, c, false, false);
    }

    // Rows M=0..7 (== batch 0..7) live in VGPRs 0..7 on lanes 0..15.
    if (lane < 16) {
        const float bias = bv[n];
#pragma unroll
        for (int r = 0; r < B_; ++r)
            out[(size_t)r * DK_ + n] = c[r] + bias;
    }
}

extern "C" void kernel_launch(void* const* d_in, const int* in_sizes, int n_in,
                              void* d_out, int out_size, void* d_ws, size_t ws_size,
                              hipStream_t stream) {
    const float* x  = (const float*)d_in[0];
    // d_in[1..4] = Wq, bq, Wk, bk -> algebraically dead, never touched.
    const float* Wv = (const float*)d_in[5];
    const float* bv = (const float*)d_in[6];
    float* out = (float*)d_out;

    float* partials = (float*)d_ws;                          // 64*8*1024 f32 (2 MB)
    float* xbarp    = partials + (size_t)NCHUNK * B_ * D_;   // 16*1024 f32 (padded)

    colsum_partial<<<dim3(B_ * NCHUNK), dim3(256), 0, stream>>>(x, partials);
    colsum_finalize_pad<<<dim3((BPAD_ * D_) / 256), dim3(256), 0, stream>>>(partials, xbarp);
    wmma_gemm_out<<<dim3(DK_ / 16), dim3(32), 0, stream>>>(xbarp, Wv, bv, out);
}